// SoftmaxMixLoss_41867341201462
// MI455X (gfx1250) — compile-verified
//
#include <hip/hip_runtime.h>
#include <stdint.h>

#define BN   4096
#define DIM  512
#define TOPK 32

typedef __attribute__((ext_vector_type(2))) float v2f;
typedef __attribute__((ext_vector_type(4))) float v4f;
typedef __attribute__((ext_vector_type(8))) float v8f;

// ---------- helpers ----------
__device__ __forceinline__ unsigned order_key(float f) {
    unsigned u = __float_as_uint(f);
    return (u & 0x80000000u) ? ~u : (u | 0x80000000u);
}
__device__ __forceinline__ float unorder_key(unsigned u) {
    unsigned b = (u & 0x80000000u) ? (u ^ 0x80000000u) : ~u;
    return __uint_as_float(b);
}
__device__ __forceinline__ float softplusf(float y) {
    return fmaxf(y, 0.f) + log1pf(expf(-fabsf(y)));
}

// ---------- 0: zero integer scratch ----------
__global__ void k_zero(int* ccount, int* ccur) {
    int i = blockIdx.x * blockDim.x + threadIdx.x;
    if (i < BN) { ccount[i] = 0; ccur[i] = 0; }
}

// ---------- 1: base-loss partial sums + off-diagonal max ----------
__global__ __launch_bounds__(256) void k_pass1(const float* __restrict__ logits,
                                               const float* __restrict__ bias,
                                               float* psum, float* pmax) {
    __shared__ float ssum[256];
    __shared__ float smax[256];
    float b = bias[0];
    float acc = 0.f, mx = -INFINITY;
    for (long long idx = (long long)blockIdx.x * blockDim.x + threadIdx.x;
         idx < (long long)BN * BN; idx += (long long)gridDim.x * blockDim.x) {
        int i = (int)(idx >> 12), j = (int)(idx & 4095);
        float x = logits[idx] + b;
        float s = (i == j) ? 1.f : -1.f;
        acc += softplusf(-s * x);          // == -log_sigmoid(s*x)
        if (i != j) mx = fmaxf(mx, x);
    }
    ssum[threadIdx.x] = acc; smax[threadIdx.x] = mx;
    __syncthreads();
    for (int s = 128; s > 0; s >>= 1) {
        if (threadIdx.x < s) {
            ssum[threadIdx.x] += ssum[threadIdx.x + s];
            smax[threadIdx.x] = fmaxf(smax[threadIdx.x], smax[threadIdx.x + s]);
        }
        __syncthreads();
    }
    if (threadIdx.x == 0) { psum[blockIdx.x] = ssum[0]; pmax[blockIdx.x] = smax[0]; }
}

__global__ __launch_bounds__(1024) void k_reduce1(const float* psum, const float* pmax, float* scal) {
    __shared__ float ss[1024];
    __shared__ float sm[1024];
    float a = 0.f, m = -INFINITY;
    for (int i = threadIdx.x; i < 4096; i += 1024) { a += psum[i]; m = fmaxf(m, pmax[i]); }
    ss[threadIdx.x] = a; sm[threadIdx.x] = m;
    __syncthreads();
    for (int s = 512; s > 0; s >>= 1) {
        if (threadIdx.x < s) {
            ss[threadIdx.x] += ss[threadIdx.x + s];
            sm[threadIdx.x] = fmaxf(sm[threadIdx.x], sm[threadIdx.x + s]);
        }
        __syncthreads();
    }
    if (threadIdx.x == 0) { scal[0] = sm[0]; scal[1] = ss[0]; }
}

// ---------- 2: top-32 per row (diag = -inf, lowest-index tiebreak) ----------
__global__ __launch_bounds__(256) void k_topk(const float* __restrict__ logits,
                                              const float* __restrict__ bias,
                                              int* __restrict__ tki) {
    __shared__ unsigned okey[BN];                 // 16 KB
    __shared__ unsigned long long red[256];       // 2 KB
    int row = blockIdx.x, t = threadIdx.x;
    float b = bias[0];
    for (int j = t; j < BN; j += 256) {
        float x = logits[(long long)row * BN + j] + b;
        okey[j] = (j == row) ? 0u : order_key(x);
    }
    __syncthreads();
    for (int it = 0; it < TOPK; ++it) {
        unsigned long long best = 0ull;
        int base = t * 16;
        #pragma unroll
        for (int q = 0; q < 16; ++q) {
            int j = base + q;
            unsigned long long cand = ((unsigned long long)okey[j] << 12) | (unsigned)(4095 - j);
            best = (cand > best) ? cand : best;
        }
        red[t] = best;
        __syncthreads();
        for (int s = 128; s > 0; s >>= 1) {
            if (t < s) { if (red[t + s] > red[t]) red[t] = red[t + s]; }
            __syncthreads();
        }
        if (t == 0) {
            int j = 4095 - (int)(red[0] & 0xFFFull);
            tki[row * TOPK + it] = j;
            okey[j] = 0u;
        }
        __syncthreads();
    }
}

// ---------- 3: kernel values (CSR) + column counts ----------
__global__ void k_kv(const float* __restrict__ logits, const float* __restrict__ bias,
                     const int* __restrict__ tki, const float* __restrict__ scal,
                     float* __restrict__ kv, int* ccount) {
    int e = blockIdx.x * blockDim.x + threadIdx.x;
    if (e >= BN * TOPK) return;
    int row = e >> 5;
    int j = tki[e];
    float x = logits[(long long)row * BN + j] + bias[0];
    // cost = max_logit - x >= 0 on mask; exp(-cost/0.05) = exp((x-max)*20)
    float kt = fmaxf(expf((x - scal[0]) * 20.0f), 1e-12f);
    kv[e] = kt;
    atomicAdd(&ccount[j], 1);
}

// ---------- 4: exclusive scan of column counts (one block) ----------
__global__ __launch_bounds__(1024) void k_scan(const int* __restrict__ ccount, int* __restrict__ coff) {
    __shared__ int part[1024];
    int t = threadIdx.x, base = t * 4;
    int loc[4], s = 0;
    for (int q = 0; q < 4; ++q) { loc[q] = s; s += ccount[base + q]; }
    part[t] = s;
    __syncthreads();
    for (int off = 1; off < 1024; off <<= 1) {
        int v = (t >= off) ? part[t - off] : 0;
        __syncthreads();
        part[t] += v;
        __syncthreads();
    }
    int pre = (t == 0) ? 0 : part[t - 1];
    for (int q = 0; q < 4; ++q) coff[base + q] = pre + loc[q];
    if (t == 1023) coff[4096] = part[1023];
}

// ---------- 5: CSC fill ----------
__global__ void k_fill(const int* __restrict__ tki, const float* __restrict__ kv,
                       const int* __restrict__ coff, int* ccur,
                       int* __restrict__ crow, float* __restrict__ cval) {
    int e = blockIdx.x * blockDim.x + threadIdx.x;
    if (e >= BN * TOPK) return;
    int row = e >> 5;
    int j = tki[e];
    int pos = coff[j] + atomicAdd(&ccur[j], 1);
    crow[pos] = row;
    cval[pos] = kv[e] - 1e-12f;     // background-adjusted value
}

// ---------- 6: sort each CSC column by row (determinism) ----------
__global__ void k_sortcols(const int* __restrict__ coff, int* __restrict__ crow, float* __restrict__ cval) {
    int j = blockIdx.x * blockDim.x + threadIdx.x;
    if (j >= BN) return;
    int lo = coff[j], hi = coff[j + 1];
    for (int a = lo + 1; a < hi; ++a) {
        int r = crow[a]; float v = cval[a];
        int b = a - 1;
        while (b >= lo && crow[b] > r) { crow[b + 1] = crow[b]; cval[b + 1] = cval[b]; --b; }
        crow[b + 1] = r; cval[b + 1] = v;
    }
}

// ---------- 7: Sinkhorn, 30 iterations, single workgroup ----------
__global__ __launch_bounds__(1024) void k_sinkhorn(const int* __restrict__ tki, const float* __restrict__ kv,
                                                   const int* __restrict__ coff, const int* __restrict__ crow,
                                                   const float* __restrict__ cval,
                                                   float* __restrict__ uvec, float* __restrict__ vvec) {
    __shared__ float su[BN];
    __shared__ float sv[BN];
    __shared__ float red[1024];
    __shared__ float bc[2];
    int t = threadIdx.x;
    for (int i = t; i < BN; i += 1024) sv[i] = 1.f;
    __syncthreads();
    for (int it = 0; it < 30; ++it) {
        float a = 0.f;
        for (int i = t; i < BN; i += 1024) a += sv[i];
        red[t] = a; __syncthreads();
        for (int s = 512; s > 0; s >>= 1) { if (t < s) red[t] += red[t + s]; __syncthreads(); }
        if (t == 0) bc[0] = red[0];
        __syncthreads();
        float sumv = bc[0];
        for (int i = t; i < BN; i += 1024) {
            float d = 1e-12f * sumv;
            const float* kr = &kv[i * TOPK];
            const int*   cr = &tki[i * TOPK];
            #pragma unroll
            for (int k = 0; k < TOPK; ++k) d += (kr[k] - 1e-12f) * sv[cr[k]];
            su[i] = (1.0f / (float)BN) / (d + 1e-8f);
        }
        __syncthreads();
        float b2 = 0.f;
        for (int i = t; i < BN; i += 1024) b2 += su[i];
        red[t] = b2; __syncthreads();
        for (int s = 512; s > 0; s >>= 1) { if (t < s) red[t] += red[t + s]; __syncthreads(); }
        if (t == 0) bc[1] = red[0];
        __syncthreads();
        float sumu = bc[1];
        for (int j = t; j < BN; j += 1024) {
            float d = 1e-12f * sumu;
            int e0 = coff[j], e1 = coff[j + 1];
            for (int e = e0; e < e1; ++e) d += cval[e] * su[crow[e]];
            sv[j] = (1.0f / (float)BN) / (d + 1e-8f);
        }
        __syncthreads();
    }
    for (int i = t; i < BN; i += 1024) { uvec[i] = su[i]; vvec[i] = sv[i]; }
}

// ---------- 8: fp32 WMMA GEMM: ratio = logits / (text @ image^T + 1e-8) ----------
// Staging uses CDNA5 async global->LDS copies (ASYNCcnt) instead of a VGPR roundtrip.
__global__ __launch_bounds__(256) void k_gemm_ratio(const float* __restrict__ text,
                                                    const float* __restrict__ img,
                                                    const float* __restrict__ logits,
                                                    float* __restrict__ ratio) {
    __shared__ float As[64][68];
    __shared__ float Bs[64][68];
    int bm = blockIdx.y, bn = blockIdx.x;
    int tid = threadIdx.x;
    int wave = tid >> 5, lane = tid & 31;
    int sc  = wave & 3;            // subtile column 0..3
    int sr0 = (wave >> 2) * 2;     // subtile rows sr0, sr0+1
    int l15 = lane & 15;
    int khalf = (lane & 16) ? 2 : 0;   // ISA 16x4 f32 A-layout: VGPR0 = K{0|2}, VGPR1 = K{1|3}
    v8f acc0 = {}; v8f acc1 = {};
    for (int kt = 0; kt < DIM; kt += 64) {
        const float* gA = text + (size_t)(bm * 64) * DIM + kt;
        const float* gB = img  + (size_t)(bn * 64) * DIM + kt;
        // 64x64 tile per matrix = 1024 16B chunks; 256 threads x 4 chunks each.
        for (int q = tid; q < 64 * 16; q += 256) {
            int r = q >> 4, c4 = (q & 15) << 2;
            unsigned voff = (unsigned)((r * DIM + c4) * 4);          // global byte offset
            unsigned dstA = (unsigned)(uintptr_t)&As[r][c4];         // LDS byte offset
            unsigned dstB = (unsigned)(uintptr_t)&Bs[r][c4];
            asm volatile("global_load_async_to_lds_b128 %0, %1, %2 offset:0"
                         :: "v"(dstA), "v"(voff), "s"(gA) : "memory");
            asm volatile("global_load_async_to_lds_b128 %0, %1, %2 offset:0"
                         :: "v"(dstB), "v"(voff), "s"(gB) : "memory");
        }
        asm volatile("s_wait_asynccnt 0" ::: "memory");
        __syncthreads();
        #pragma unroll
        for (int kk = 0; kk < 64; kk += 4) {
            int kc = kk + khalf;
            v2f a0, a1, bb;
            a0.x = As[sr0 * 16 + l15][kc];       a0.y = As[sr0 * 16 + l15][kc + 1];
            a1.x = As[sr0 * 16 + 16 + l15][kc];  a1.y = As[sr0 * 16 + 16 + l15][kc + 1];
            bb.x = Bs[sc * 16 + l15][kc];        bb.y = Bs[sc * 16 + l15][kc + 1];
            acc0 = __builtin_amdgcn_wmma_f32_16x16x4_f32(false, a0, false, bb, (short)0, acc0, false, false);
            acc1 = __builtin_amdgcn_wmma_f32_16x16x4_f32(false, a1, false, bb, (short)0, acc1, false, false);
        }
        __syncthreads();
    }
    int colg = bn * 64 + sc * 16 + l15;
    int r0 = bm * 64 + sr0 * 16 + ((lane & 16) ? 8 : 0);
    int r1 = r0 + 16;
    #pragma unroll
    for (int r = 0; r < 8; ++r) {
        long long i0 = (long long)(r0 + r) * BN + colg;
        ratio[i0] = logits[i0] / (acc0[r] + 1e-8f);
        long long i1 = (long long)(r1 + r) * BN + colg;
        ratio[i1] = logits[i1] / (acc1[r] + 1e-8f);
    }
}

// ---------- 9: radix-select median (two order statistics) ----------
__global__ void k_selinit(unsigned* sel, unsigned* hist, unsigned rank) {
    int t = threadIdx.x;
    if (t == 0) { sel[0] = 0u; sel[1] = rank; }
    if (t < 256) hist[t] = 0u;
}

__global__ __launch_bounds__(256) void k_hist(const float* __restrict__ ratio,
                                              const unsigned* __restrict__ sel,
                                              unsigned* __restrict__ hist, int level) {
    __shared__ unsigned lh[256];
    if (threadIdx.x < 256) lh[threadIdx.x] = 0u;
    __syncthreads();
    unsigned pref = sel[0];
    int shift = 24 - 8 * level;
    long long stride = (long long)gridDim.x * blockDim.x;
    for (long long idx = (long long)blockIdx.x * blockDim.x + threadIdx.x;
         idx < (long long)BN * BN; idx += stride) {
        if (idx + stride < (long long)BN * BN) __builtin_prefetch(&ratio[idx + stride], 0, 0);
        int i = (int)(idx >> 12), j = (int)(idx & 4095);
        if (i == j) continue;
        unsigned key = order_key(ratio[idx]);
        bool ok = (level == 0) || ((key >> (shift + 8)) == (pref >> (shift + 8)));
        if (ok) atomicAdd(&lh[(key >> shift) & 255u], 1u);
    }
    __syncthreads();
    if (threadIdx.x < 256 && lh[threadIdx.x]) atomicAdd(&hist[threadIdx.x], lh[threadIdx.x]);
}

__global__ void k_pick(unsigned* sel, unsigned* hist, float* scal, int level, int slot) {
    int t = threadIdx.x;
    if (t == 0) {
        int shift = 24 - 8 * level;
        unsigned r = sel[1], cum = 0;
        unsigned pref = sel[0];
        for (int b = 0; b < 256; ++b) {
            unsigned h = hist[b];
            if (cum + h > r) { pref |= ((unsigned)b) << shift; sel[1] = r - cum; break; }
            cum += h;
        }
        sel[0] = pref;
        if (level == 3) scal[2 + slot] = unorder_key(pref);
    }
    __syncthreads();
    if (t < 256) hist[t] = 0u;
}

// ---------- 10: synthetic negative similarity per row ----------
__global__ __launch_bounds__(256) void k_synth(const float* __restrict__ text,
                                               const float* __restrict__ img,
                                               const int* __restrict__ tki,
                                               const float* __restrict__ kv,
                                               const float* __restrict__ uvec,
                                               const float* __restrict__ vvec,
                                               float* __restrict__ ssim) {
    __shared__ float w[TOPK];
    __shared__ float red[256];
    __shared__ float inv_mass;
    __shared__ float nrm;
    int row = blockIdx.x, t = threadIdx.x;
    if (t < TOPK) {
        int j = tki[row * TOPK + t];
        w[t] = uvec[row] * kv[row * TOPK + t] * vvec[j];   // plan entries on mask
    }
    __syncthreads();
    if (t == 0) {
        float s = 0.f;
        for (int k = 0; k < TOPK; ++k) s += w[k];
        inv_mass = 1.0f / fmaxf(s, 1e-8f);
    }
    __syncthreads();
    int d0 = t, d1 = t + 256;
    float s0 = 0.f, s1 = 0.f;
    for (int k = 0; k < TOPK; ++k) {
        int j = tki[row * TOPK + k];
        float ww = w[k] * inv_mass;
        s0 += ww * img[(size_t)j * DIM + d0];
        s1 += ww * img[(size_t)j * DIM + d1];
    }
    red[t] = s0 * s0 + s1 * s1;
    __syncthreads();
    for (int s = 128; s > 0; s >>= 1) { if (t < s) red[t] += red[t + s]; __syncthreads(); }
    if (t == 0) nrm = sqrtf(red[0]);
    __syncthreads();
    red[t] = s0 * text[(size_t)row * DIM + d0] + s1 * text[(size_t)row * DIM + d1];
    __syncthreads();
    for (int s = 128; s > 0; s >>= 1) { if (t < s) red[t] += red[t + s]; __syncthreads(); }
    if (t == 0) ssim[row] = red[0] / (nrm + 1e-8f);
}

// ---------- 11: final gate + combine ----------
__global__ __launch_bounds__(512) void k_final(const float* __restrict__ ssim,
                                               const float* __restrict__ scal,
                                               float* __restrict__ out) {
    __shared__ float rg[512];
    __shared__ float rl[512];
    int t = threadIdx.x;
    float scale = 0.5f * (scal[2] + scal[3]);     // median = avg of two middle order stats
    float g = 0.f, l = 0.f;
    for (int i = t; i < BN; i += 512) {
        float z = scale * ssim[i];
        float gate = (z > -0.05f) ? 1.f : 0.f;
        l += softplusf(z) * gate;                 // -log_sigmoid(-z) = softplus(z)
        g += gate;
    }
    rg[t] = g; rl[t] = l;
    __syncthreads();
    for (int s = 256; s > 0; s >>= 1) { if (t < s) { rg[t] += rg[t + s]; rl[t] += rl[t + s]; } __syncthreads(); }
    if (t == 0) {
        float gs = rg[0];
        float synth = (gs > 0.f) ? (rl[0] / (gs + 1e-8f)) : 0.f;
        out[0] = scal[1] / ((float)BN * (float)BN) + 0.5f * synth;
    }
}

// ---------- launcher ----------
extern "C" void kernel_launch(void* const* d_in, const int* in_sizes, int n_in,
                              void* d_out, int out_size, void* d_ws, size_t ws_size,
                              hipStream_t stream) {
    (void)in_sizes; (void)n_in; (void)out_size; (void)ws_size;
    const float* logits = (const float*)d_in[0];
    const float* text   = (const float*)d_in[1];
    const float* img    = (const float*)d_in[2];
    const float* bias   = (const float*)d_in[3];
    float* out = (float*)d_out;

    char* p = (char*)d_ws;
    float* ratio = (float*)p;              p += (size_t)BN * BN * 4;
    int*   tki   = (int*)p;                p += (size_t)BN * TOPK * 4;
    float* kv    = (float*)p;              p += (size_t)BN * TOPK * 4;
    int*   ccount= (int*)p;                p += (size_t)BN * 4;
    int*   coff  = (int*)p;                p += (size_t)(BN + 8) * 4;
    int*   ccur  = (int*)p;                p += (size_t)BN * 4;
    int*   crow  = (int*)p;                p += (size_t)BN * TOPK * 4;
    float* cval  = (float*)p;              p += (size_t)BN * TOPK * 4;
    float* uvec  = (float*)p;              p += (size_t)BN * 4;
    float* vvec  = (float*)p;              p += (size_t)BN * 4;
    float* ssim  = (float*)p;              p += (size_t)BN * 4;
    float* psum  = (float*)p;              p += 4096 * 4;
    float* pmax  = (float*)p;              p += 4096 * 4;
    float* scal  = (float*)p;              p += 32 * 4;
    unsigned* hist = (unsigned*)p;         p += 256 * 4;
    unsigned* sel  = (unsigned*)p;         p += 8 * 4;

    k_zero<<<16, 256, 0, stream>>>(ccount, ccur);
    k_pass1<<<4096, 256, 0, stream>>>(logits, bias, psum, pmax);
    k_reduce1<<<1, 1024, 0, stream>>>(psum, pmax, scal);
    k_topk<<<4096, 256, 0, stream>>>(logits, bias, tki);
    k_kv<<<(BN * TOPK) / 256, 256, 0, stream>>>(logits, bias, tki, scal, kv, ccount);
    k_scan<<<1, 1024, 0, stream>>>(ccount, coff);
    k_fill<<<(BN * TOPK) / 256, 256, 0, stream>>>(tki, kv, coff, ccur, crow, cval);
    k_sortcols<<<16, 256, 0, stream>>>(coff, crow, cval);
    k_sinkhorn<<<1, 1024, 0, stream>>>(tki, kv, coff, crow, cval, uvec, vvec);

    dim3 ggrid(64, 64);
    k_gemm_ratio<<<ggrid, 256, 0, stream>>>(text, img, logits, ratio);

    k_synth<<<4096, 256, 0, stream>>>(text, img, tki, kv, uvec, vvec, ssim);

    // median = avg of order statistics N/2-1 and N/2 over N = BN*(BN-1) off-diagonal entries
    const long long NOFF = (long long)BN * (BN - 1);
    unsigned ranks[2] = { (unsigned)(NOFF / 2 - 1), (unsigned)(NOFF / 2) };
    for (int slot = 0; slot < 2; ++slot) {
        k_selinit<<<1, 256, 0, stream>>>(sel, hist, ranks[slot]);
        for (int lvl = 0; lvl < 4; ++lvl) {
            k_hist<<<2048, 256, 0, stream>>>(ratio, sel, hist, lvl);
            k_pick<<<1, 256, 0, stream>>>(sel, hist, scal, lvl, slot);
        }
    }

    k_final<<<1, 512, 0, stream>>>(ssim, scal, out);
}